// ModelNew_17514876633576
// MI455X (gfx1250) — compile-verified
//
#include <hip/hip_runtime.h>
#include <hip/hip_bf16.h>

// ---------------------------------------------------------------------------
// Problem: out[b] = dot(x[b, :], colsum(W)) + sum(bias)
//   x: (65536, 4096) f32, W: (4096, 4096) f32, bias: (4096,) f32
// HBM-bound: 1.07 GB of x dominates (~46 us @ 23.3 TB/s).
// GEMV mapped onto V_WMMA_F32_16X16X4_F32 (f32 in / f32 acc, wave32), with
// double-buffered GLOBAL_LOAD_ASYNC_TO_LDS_B128 staging (ASYNCcnt pipeline).
// ---------------------------------------------------------------------------

typedef float v8f __attribute__((ext_vector_type(8)));
typedef float v2f __attribute__((ext_vector_type(2)));

#define BATCH        65536
#define IN_F         4096
#define OUT_F        4096
#define ROWS_PER_BLK 128      // 8 waves x 16 rows
#define KCHUNK       64       // K floats staged per chunk
#define LDS_STRIDE   68       // 64 + 4 pad; 68 % 64 == 4 -> conflict-free A reads
#define ROW_GROUPS   32       // colsum partial row groups (128 rows each)

// --------------------------- Phase A1: W column partial sums ----------------
__global__ __launch_bounds__(256) void colsum_partial_kernel(
    const float* __restrict__ w, float* __restrict__ partial) {
  const int col = blockIdx.x * 256 + threadIdx.x;           // 0..4095
  const long rb = (long)blockIdx.y * (OUT_F / ROW_GROUPS);  // 128 rows / group
  float acc = 0.0f;
#pragma unroll 8
  for (int i = 0; i < OUT_F / ROW_GROUPS; ++i)
    acc += w[(rb + i) * IN_F + col];
  partial[blockIdx.y * IN_F + col] = acc;
}

// --------------------------- Phase A2: reduce partials ----------------------
__global__ __launch_bounds__(256) void colsum_final_kernel(
    const float* __restrict__ partial, float* __restrict__ v) {
  const int col = blockIdx.x * 256 + threadIdx.x;
  float acc = 0.0f;
#pragma unroll
  for (int g = 0; g < ROW_GROUPS; ++g) acc += partial[g * IN_F + col];
  v[col] = acc;
}

// --------------------------- Phase A3: bias sum -----------------------------
__global__ __launch_bounds__(256) void bias_sum_kernel(
    const float* __restrict__ bias, float* __restrict__ bsum) {
  __shared__ float red[256];
  float acc = 0.0f;
  for (int i = threadIdx.x; i < OUT_F; i += 256) acc += bias[i];
  red[threadIdx.x] = acc;
  __syncthreads();
  for (int s = 128; s > 0; s >>= 1) {
    if (threadIdx.x < (unsigned)s) red[threadIdx.x] += red[threadIdx.x + s];
    __syncthreads();
  }
  if (threadIdx.x == 0) *bsum = red[0];
}

// --------------------------- Phase B: GEMV via WMMA f32 ---------------------
// Each wave: 16 rows, D += A(16x4) * B(4x16) with B = broadcast v[k..k+3].
// A fragment (ISA 7.12.2, 32-bit A 16x4): lane L -> row (L&15), VGPR0/1 hold
//   K = 2*(L>>4), 2*(L>>4)+1.  B fragment: VGPR0/1 lane L hold
//   v[k + 2*(L>>4)], v[k + 2*(L>>4)+1] replicated over all 16 N columns.
// Staging: double-buffered GLOBAL_LOAD_ASYNC_TO_LDS_B128; each thread issues
// 8 async b128 loads per chunk; async loads complete in order, so
// s_wait_asynccnt 0x8 after issuing the next chunk proves the current chunk
// is resident; the barrier publishes it block-wide.
__global__ __launch_bounds__(256) void gemv_wmma_kernel(
    const float* __restrict__ x, const float* __restrict__ v,
    const float* __restrict__ bsum, float* __restrict__ out) {
  __shared__ float xs[2][ROWS_PER_BLK * LDS_STRIDE];  // 2 x 34,816 B

  const int tid    = threadIdx.x;
  const int lane   = tid & 31;
  const int wave   = tid >> 5;        // 0..7
  const int laneHi = lane >> 4;       // 0 or 1
  const int rowInW = lane & 15;       // A-fragment row within wave tile

  const long rowBase = (long)blockIdx.x * ROWS_PER_BLK;

  // staging geometry: 16 threads per row, 4 floats each -> 16 rows per pass
  const int sRow = tid >> 4;           // 0..15
  const int sCol = (tid & 15) * 4;     // 0..60

  const float* gbase = x + rowBase * IN_F + (long)sRow * IN_F + sCol;
  const unsigned lds0 =
      (unsigned)(unsigned long long)(&xs[0][sRow * LDS_STRIDE + sCol]);
  const unsigned lds1 =
      (unsigned)(unsigned long long)(&xs[1][sRow * LDS_STRIDE + sCol]);

  // ---- prologue: async-stage chunk 0 into buffer 0 ----
#pragma unroll
  for (int rr = 0; rr < 8; ++rr) {
    const float* g = gbase + (long)rr * 16 * IN_F;        // kc = 0
    const unsigned l = lds0 + rr * 16 * LDS_STRIDE * 4;
    asm volatile("global_load_async_to_lds_b128 %0, %1, off"
                 :: "v"(l), "v"(g) : "memory");
  }

  v8f acc = {};
  int buf = 0;

  for (int kc = 0; kc < IN_F; kc += KCHUNK) {
    if (kc + KCHUNK < IN_F) {
      // ---- async-stage next chunk into the other buffer ----
      const unsigned ldsn = (buf ? lds0 : lds1);
#pragma unroll
      for (int rr = 0; rr < 8; ++rr) {
        const float* g = gbase + (long)rr * 16 * IN_F + (kc + KCHUNK);
        const unsigned l = ldsn + rr * 16 * LDS_STRIDE * 4;
        asm volatile("global_load_async_to_lds_b128 %0, %1, off"
                     :: "v"(l), "v"(g) : "memory");
      }
      asm volatile("s_wait_asynccnt 0x8" ::: "memory");   // current chunk done
    } else {
      asm volatile("s_wait_asynccnt 0x0" ::: "memory");   // last chunk done
    }
    __syncthreads();

    const float* xw = &xs[buf][(wave * 16 + rowInW) * LDS_STRIDE + laneHi * 2];
#pragma unroll
    for (int i = 0; i < KCHUNK; i += 4) {
      v2f a = *(const v2f*)(xw + i);                    // ds_load_b64, bank-clean
      const float4 vv = *(const float4*)(v + kc + i);   // wave-uniform -> s_load
      v2f b;
      b.x = laneHi ? vv.z : vv.x;
      b.y = laneHi ? vv.w : vv.y;
      // 8 args: (neg_a, A, neg_b, B, c_mod, C, reuse_a, reuse_b)
      acc = __builtin_amdgcn_wmma_f32_16x16x4_f32(
          false, a, false, b, (short)0, acc, false, false);
    }
    __syncthreads();
    buf ^= 1;
  }

  // All N columns of D are identical; lane 0 holds rows tile+0..7 in acc[0..7],
  // lane 16 holds rows tile+8..15.
  const float bs = *bsum;
  if ((lane & 15) == 0) {
    float* o = out + rowBase + wave * 16 + laneHi * 8;
#pragma unroll
    for (int j = 0; j < 8; ++j) o[j] = acc[j] + bs;
  }
}

// ---------------------------------------------------------------------------
extern "C" void kernel_launch(void* const* d_in, const int* in_sizes, int n_in,
                              void* d_out, int out_size, void* d_ws, size_t ws_size,
                              hipStream_t stream) {
  (void)in_sizes; (void)n_in; (void)out_size; (void)ws_size;

  const float* x    = (const float*)d_in[0];  // 65536 x 4096
  const float* w    = (const float*)d_in[1];  // 4096 x 4096
  const float* bias = (const float*)d_in[2];  // 4096
  float*       out  = (float*)d_out;          // 65536

  float* ws      = (float*)d_ws;
  float* v       = ws;              // 4096 floats: colsum(W)
  float* bsum    = ws + IN_F;       // 1 float
  float* partial = ws + 2 * IN_F;   // 32 * 4096 floats

  colsum_partial_kernel<<<dim3(IN_F / 256, ROW_GROUPS), 256, 0, stream>>>(w, partial);
  colsum_final_kernel<<<dim3(IN_F / 256), 256, 0, stream>>>(partial, v);
  bias_sum_kernel<<<1, 256, 0, stream>>>(bias, bsum);
  gemv_wmma_kernel<<<dim3(BATCH / ROWS_PER_BLK), 256, 0, stream>>>(x, v, bsum, out);
}